// ModelNew_25056839205217
// MI455X (gfx1250) — compile-verified
//
#include <hip/hip_runtime.h>

typedef float v2f __attribute__((ext_vector_type(2)));
typedef float v8f __attribute__((ext_vector_type(8)));

#define NMAT 4096
#define BM 128
#define BN 64
#define BK 16
#define KS (BK + 4)  // 20-float row stride: 16B-aligned rows, conflict-free frag reads

__global__ __launch_bounds__(256) void tril_gemm_wmma_f32(
    const float* __restrict__ A, const float* __restrict__ B,
    float* __restrict__ C) {
  // M-major tiles, double buffered: fragment = contiguous K pair -> ds_load_b64
  __shared__ float As[2][BM][KS];
  __shared__ float Bs[2][BN][KS];

  const int tid = threadIdx.x;
  const int wid = tid >> 5;
  const int lane = tid & 31;
  const int lane16 = lane & 15;
  const int hi = lane >> 4;  // half-wave selects K pair (K = 2*hi + v)

  const int rowStart = blockIdx.y * BM;
  const int colStart = blockIdx.x * BN;

  const int waveM = (wid & 3) * 32;   // 4 waves down
  const int waveN = (wid >> 2) * 32;  // 2 waves across

  // A staging map: 2 x float4 along K per thread
  // B staging map: one column n, 4 consecutive k per thread
  const int bn = tid & 63;
  const int bkq = tid >> 6;  // 0..3 -> k quad

  v8f acc[2][2] = {};
  float4 ra[2];
  float rb[4];

  // Effective k-range: B mask needs k >= j >= colStart; A mask needs k <= i < rowStart+BM.
  const int kStart = colStart;
  const int kEnd = rowStart + BM;

  auto loadTile = [&](int kt) {
#pragma unroll
    for (int r = 0; r < 2; ++r) {
      const int idx = tid + r * 256;
      const int m = idx >> 2;
      const int cv = idx & 3;
      ra[r] = *reinterpret_cast<const float4*>(
          &A[(size_t)(rowStart + m) * NMAT + kt + cv * 4]);
    }
#pragma unroll
    for (int e = 0; e < 4; ++e) {
      rb[e] = B[(size_t)(kt + bkq * 4 + e) * NMAT + colStart + bn];
    }
    // speculative prefetch one more tile ahead
    if (kt + BK < kEnd) {
      __builtin_prefetch(&A[(size_t)(rowStart + (tid >> 1)) * NMAT + kt + BK],
                         0, 1);
    }
  };

  auto stageTile = [&](int kt, int buf) {
#pragma unroll
    for (int r = 0; r < 2; ++r) {
      const int idx = tid + r * 256;
      const int m = idx >> 2;
      const int cv = idx & 3;
      const int gi = rowStart + m;
      const int gk = kt + cv * 4;
      const float t[4] = {ra[r].x, ra[r].y, ra[r].z, ra[r].w};
      float4 s;
      s.x = (gk + 0 <= gi) ? t[0] : 0.0f;
      s.y = (gk + 1 <= gi) ? t[1] : 0.0f;
      s.z = (gk + 2 <= gi) ? t[2] : 0.0f;
      s.w = (gk + 3 <= gi) ? t[3] : 0.0f;
      *reinterpret_cast<float4*>(&As[buf][m][cv * 4]) = s;  // ds_store_b128
    }
    {
      const int gj = colStart + bn;
      float4 s;
      s.x = (gj <= kt + bkq * 4 + 0) ? rb[0] : 0.0f;
      s.y = (gj <= kt + bkq * 4 + 1) ? rb[1] : 0.0f;
      s.z = (gj <= kt + bkq * 4 + 2) ? rb[2] : 0.0f;
      s.w = (gj <= kt + bkq * 4 + 3) ? rb[3] : 0.0f;
      *reinterpret_cast<float4*>(&Bs[buf][bn][bkq * 4]) = s;  // ds_store_b128
    }
  };

  auto compute = [&](int buf) {
#pragma unroll
    for (int kk = 0; kk < BK; kk += 4) {
      v2f a[2], b[2];
#pragma unroll
      for (int mi = 0; mi < 2; ++mi) {
        a[mi] = *reinterpret_cast<const v2f*>(
            &As[buf][waveM + mi * 16 + lane16][kk + 2 * hi]);  // ds_load_b64
      }
#pragma unroll
      for (int ni = 0; ni < 2; ++ni) {
        b[ni] = *reinterpret_cast<const v2f*>(
            &Bs[buf][waveN + ni * 16 + lane16][kk + 2 * hi]);  // ds_load_b64
      }
#pragma unroll
      for (int mi = 0; mi < 2; ++mi) {
#pragma unroll
        for (int ni = 0; ni < 2; ++ni) {
          acc[mi][ni] = __builtin_amdgcn_wmma_f32_16x16x4_f32(
              false, a[mi], false, b[ni], (short)0, acc[mi][ni], false, false);
        }
      }
    }
  };

  // -------- software-pipelined main loop (double-buffered LDS) --------
  int buf = 0;
  if (kStart < kEnd) {
    loadTile(kStart);
    stageTile(kStart, 0);
  }
  for (int kt = kStart; kt < kEnd; kt += BK) {
    __syncthreads();
    const bool hasNext = (kt + BK) < kEnd;  // block-uniform
    if (hasNext) loadTile(kt + BK);         // global loads in flight...
    compute(buf);                           // ...overlap with WMMA from LDS
    if (hasNext) stageTile(kt + BK, buf ^ 1);
    buf ^= 1;
  }

  // ---- Store with lower-triangular mask, diagonal kept (zeroes strict upper triangle) ----
#pragma unroll
  for (int mi = 0; mi < 2; ++mi) {
#pragma unroll
    for (int ni = 0; ni < 2; ++ni) {
      const int gj = colStart + waveN + ni * 16 + lane16;
#pragma unroll
      for (int r = 0; r < 8; ++r) {
        const int gi = rowStart + waveM + mi * 16 + r + 8 * hi;
        const float v = (gi >= gj) ? acc[mi][ni][r] : 0.0f;
        C[(size_t)gi * NMAT + gj] = v;
      }
    }
  }
}

extern "C" void kernel_launch(void* const* d_in, const int* in_sizes, int n_in,
                              void* d_out, int out_size, void* d_ws,
                              size_t ws_size, hipStream_t stream) {
  (void)in_sizes; (void)n_in; (void)out_size; (void)d_ws; (void)ws_size;
  const float* A = (const float*)d_in[0];
  const float* B = (const float*)d_in[1];
  float* C = (float*)d_out;
  dim3 grid(NMAT / BN, NMAT / BM);  // (64, 32)
  tril_gemm_wmma_f32<<<grid, dim3(256), 0, stream>>>(A, B, C);
}